// ICPEstimator_16320875725009
// MI455X (gfx1250) — compile-verified
//
#include <hip/hip_runtime.h>

typedef __attribute__((ext_vector_type(2))) float v2f;
typedef __attribute__((ext_vector_type(8))) float v8f;

#define TAU_LM 0.001f
#define NPTS   16384            // fixed by the reference (N = 16384)

// ---- workspace layout (float offsets) ----
#define WS_X     0     // accepted x[6]
#define WS_F     6     // accepted F
#define WS_MU    7
#define WS_NU    8
#define WS_HP    9     // accepted packed H[21] (upper-tri row major)
#define WS_G     30    // accepted g[6]
#define WS_XNEW  36    // candidate x[6]
#define WS_DENOM 42
#define WS_HN    43    // candidate packed H[21]
#define WS_GN    64    // candidate g[6]
#define WS_FN    70    // candidate F
#define WS_R     71    // R[9] row-major for current eval pose
#define WS_T     80    // t[3]
#define WS_PART  96    // block partials: nblocks*28 floats
#define WS_TRGT  2048  // target SoA planes: [x | y | z | |t|^2], each NPTS floats
#define WS_IDS   (WS_TRGT + 4 * NPTS)   // NN ids, NPTS ints
// total: ~84K floats = ~336 KB of d_ws

__device__ __forceinline__ int hidx(int a, int b) {
  if (a > b) { int t = a; a = b; b = t; }
  return a * 6 + b - (a * (a + 1)) / 2;   // diag: 0,6,11,15,18,20
}

__device__ void se3_exp_dev(const float* x, float* R, float* t) {
  float w0 = x[0], w1 = x[1], w2 = x[2];
  float v0 = x[3], v1 = x[4], v2 = x[5];
  float th2 = w0 * w0 + w1 * w1 + w2 * w2;
  float th = sqrtf(th2);
  float A, B, C;
  if (th < 1e-8f) {
    A = 1.0f - th2 / 6.0f;
    B = 0.5f - th2 / 24.0f;
    C = 1.0f / 6.0f - th2 / 120.0f;
  } else {
    float s = sinf(th), c = cosf(th);
    A = s / th;
    B = (1.0f - c) / th2;
    C = (th - s) / (th2 * th);
  }
  float W[9]  = {0.f, -w2, w1,  w2, 0.f, -w0,  -w1, w0, 0.f};
  float W2[9];
  for (int i = 0; i < 3; ++i)
    for (int j = 0; j < 3; ++j) {
      float s = 0.f;
      for (int k = 0; k < 3; ++k) s += W[i * 3 + k] * W[k * 3 + j];
      W2[i * 3 + j] = s;
    }
  float V[9];
  for (int i = 0; i < 3; ++i)
    for (int j = 0; j < 3; ++j) {
      float I = (i == j) ? 1.0f : 0.0f;
      R[i * 3 + j] = I + A * W[i * 3 + j] + B * W2[i * 3 + j];
      V[i * 3 + j] = I + B * W[i * 3 + j] + C * W2[i * 3 + j];
    }
  t[0] = V[0] * v0 + V[1] * v1 + V[2] * v2;
  t[1] = V[3] * v0 + V[4] * v1 + V[5] * v2;
  t[2] = V[6] * v0 + V[7] * v1 + V[8] * v2;
}

// ---- init: build target SoA planes (x,y,z,|t|^2) and x_new = 0 ----
__global__ void icp_init_kernel(const float* __restrict__ trg, float* __restrict__ ws) {
  int i = blockIdx.x * blockDim.x + threadIdx.x;
  if (i < NPTS) {
    float a = trg[3 * i], b = trg[3 * i + 1], c = trg[3 * i + 2];
    float* T = ws + WS_TRGT;
    T[0 * NPTS + i] = a;
    T[1 * NPTS + i] = b;
    T[2 * NPTS + i] = c;
    T[3 * NPTS + i] = a * a + b * b + c * c;
  }
  if (i < 6) ws[WS_XNEW + i] = 0.0f;
}

// ---- pose: R,t = se3_exp(x_new) ----
__global__ void icp_pose_kernel(float* __restrict__ ws) {
  if (threadIdx.x == 0) se3_exp_dev(ws + WS_XNEW, ws + WS_R, ws + WS_T);
}

// ---- NN search with f32 WMMA 16x16x4, C = inline 0 ----
// A row i = (-2*rt0, -2*rt1, -2*rt2, 1) ;  B col j = (x_j, y_j, z_j, |t_j|^2)
// => D[i][j] = |trg_j|^2 - 2 * ref_t_i . trg_j   (row-constant |ref_t|^2 dropped)
// block = 256 threads (8 waves), 16 rows per wave -> 128 rows per block
__global__ void icp_nn_kernel(const float* __restrict__ ref, float* __restrict__ ws) {
  const float* Rm   = ws + WS_R;
  const float* tv   = ws + WS_T;
  const float* trgT = ws + WS_TRGT;
  int* ids = (int*)(ws + WS_IDS);

  int lane = threadIdx.x & 31;
  int wave = threadIdx.x >> 5;
  int rb   = blockIdx.x * 128 + wave * 16;
  int m    = lane & 15;
  int half = lane >> 4;
  int k0   = 2 * half;            // A/B K index split across lane halves

  float R0 = Rm[0], R1 = Rm[1], R2 = Rm[2];
  float R3 = Rm[3], R4 = Rm[4], R5 = Rm[5];
  float R6 = Rm[6], R7 = Rm[7], R8 = Rm[8];
  float t0 = tv[0], t1 = tv[1], t2 = tv[2];

  // ref_t = R^T (ref - t) for row rb+m
  int ri = rb + m;
  float px = ref[3 * ri]     - t0;
  float py = ref[3 * ri + 1] - t1;
  float pz = ref[3 * ri + 2] - t2;
  float a0 = -2.0f * (R0 * px + R3 * py + R6 * pz);
  float a1 = -2.0f * (R1 * px + R4 * py + R7 * pz);
  float a2 = -2.0f * (R2 * px + R5 * py + R8 * pz);

  v2f a;
  a.x = half ? a2   : a0;   // K = k0
  a.y = half ? 1.0f : a1;   // K = k0+1 ; K=3 multiplies the |trg|^2 plane

  // per-lane plane base index; second plane is a compile-time +NPTS immediate
  int off0 = k0 * NPTS + m;

  float best[8];
  int   bidx[8];
#pragma unroll
  for (int v = 0; v < 8; ++v) { best[v] = 3.4e38f; bidx[v] = 0; }

  // unroll by 2: both tiles' loads issue before either WMMA consumes them
  for (int jb = 0; jb < NPTS; jb += 32) {
    int i0 = off0 + jb;
    v2f b0, b1;
    b0.x = trgT[i0];
    b0.y = trgT[i0 + NPTS];
    b1.x = trgT[i0 + 16];
    b1.y = trgT[i0 + 16 + NPTS];

    v8f cz = {};
    v8f d0 = __builtin_amdgcn_wmma_f32_16x16x4_f32(false, a, false, b0, (short)0, cz,
                                                   false, false);
    v8f d1 = __builtin_amdgcn_wmma_f32_16x16x4_f32(false, a, false, b1, (short)0, cz,
                                                   false, false);
    int j0 = jb + m;
#pragma unroll
    for (int v = 0; v < 8; ++v) {
      float dv = d0[v];
      if (dv < best[v]) { best[v] = dv; bidx[v] = j0; }
    }
#pragma unroll
    for (int v = 0; v < 8; ++v) {
      float dv = d1[v];
      if (dv < best[v]) { best[v] = dv; bidx[v] = j0 + 16; }
    }
  }

  // argmin across the 16 lanes holding each row (xor stays within half-wave)
#pragma unroll
  for (int v = 0; v < 8; ++v) {
    float bv = best[v];
    int   bi = bidx[v];
#pragma unroll
    for (int off = 8; off >= 1; off >>= 1) {
      float ov = __shfl_xor(bv, off, 32);
      int   oi = __shfl_xor(bi, off, 32);
      if (ov < bv || (ov == bv && oi < bi)) { bv = ov; bi = oi; }
    }
    if (m == 0) ids[rb + v + 8 * half] = bi;   // C/D row = v + 8*half
  }
}

// ---- residual/Jacobian + block-level partial reduction of H(21), g(6), F(1) ----
__global__ void icp_resjac_partial(const float* __restrict__ ref, const float* __restrict__ trg,
                                   const float* __restrict__ nrm, float* __restrict__ ws) {
  __shared__ float sh[8][28];
  int i    = blockIdx.x * 256 + threadIdx.x;
  int lane = threadIdx.x & 31;
  int wave = threadIdx.x >> 5;

  const float* Rm = ws + WS_R;
  const float* tv = ws + WS_T;
  const int* ids  = (const int*)(ws + WS_IDS);

  float R0 = Rm[0], R1 = Rm[1], R2 = Rm[2];
  float R3 = Rm[3], R4 = Rm[4], R5 = Rm[5];
  float R6 = Rm[6], R7 = Rm[7], R8 = Rm[8];
  float t0 = tv[0], t1 = tv[1], t2 = tv[2];

  int id = ids[i];
  float q0 = trg[3 * id], q1 = trg[3 * id + 1], q2 = trg[3 * id + 2];
  float m0 = nrm[3 * id], m1 = nrm[3 * id + 1], m2 = nrm[3 * id + 2];
  // p = R*q + t ; n = R*m
  float p0 = R0 * q0 + R1 * q1 + R2 * q2 + t0;
  float p1 = R3 * q0 + R4 * q1 + R5 * q2 + t1;
  float p2 = R6 * q0 + R7 * q1 + R8 * q2 + t2;
  float n0 = R0 * m0 + R1 * m1 + R2 * m2;
  float n1 = R3 * m0 + R4 * m1 + R5 * m2;
  float n2 = R6 * m0 + R7 * m1 + R8 * m2;

  float rx = ref[3 * i], ry = ref[3 * i + 1], rz = ref[3 * i + 2];
  float r = n0 * (rx - p0) + n1 * (ry - p1) + n2 * (rz - p2);
  float J[6];
  J[0] = ry * n2 - rz * n1;   // cross(ref, n)
  J[1] = rz * n0 - rx * n2;
  J[2] = rx * n1 - ry * n0;
  J[3] = n0; J[4] = n1; J[5] = n2;

  float acc[28];
  int c = 0;
  for (int aa = 0; aa < 6; ++aa)
    for (int bb = aa; bb < 6; ++bb) acc[c++] = J[aa] * J[bb];
  for (int aa = 0; aa < 6; ++aa) acc[c++] = J[aa] * r;
  acc[27] = r * r;

#pragma unroll
  for (int k = 0; k < 28; ++k) {
#pragma unroll
    for (int off = 16; off >= 1; off >>= 1)
      acc[k] += __shfl_xor(acc[k], off, 32);
  }
  if (lane == 0)
    for (int k = 0; k < 28; ++k) sh[wave][k] = acc[k];
  __syncthreads();
  if (threadIdx.x < 28) {
    float s = 0.0f;
    for (int w = 0; w < 8; ++w) s += sh[w][threadIdx.x];
    ws[WS_PART + blockIdx.x * 28 + threadIdx.x] = s;
  }
}

// ---- deterministic final reduction -> Hn, gn, Fn ----
__global__ void icp_resjac_final(float* __restrict__ ws, int nblocks) {
  int c = threadIdx.x;
  if (c < 28) {
    float s = 0.0f;
    for (int b = 0; b < nblocks; ++b) s += ws[WS_PART + b * 28 + c];
    if (c < 21)      ws[WS_HN + c] = s;
    else if (c < 27) ws[WS_GN + c - 21] = s;
    else             ws[WS_FN] = s;
  }
}

// ---- LM start: accept initial eval, mu = TAU*max(diag H0), nu = 2 ----
__global__ void icp_lm_start(float* __restrict__ ws) {
  if (threadIdx.x == 0) {
    for (int k = 0; k < 6; ++k)  ws[WS_X + k] = ws[WS_XNEW + k];
    for (int k = 0; k < 21; ++k) ws[WS_HP + k] = ws[WS_HN + k];
    for (int k = 0; k < 6; ++k)  ws[WS_G + k] = ws[WS_GN + k];
    ws[WS_F] = ws[WS_FN];
    const int dg[6] = {0, 6, 11, 15, 18, 20};
    float mx = ws[WS_HN + dg[0]];
    for (int k = 1; k < 6; ++k) mx = fmaxf(mx, ws[WS_HN + dg[k]]);
    ws[WS_MU] = TAU_LM * mx;
    ws[WS_NU] = 2.0f;
  }
}

// ---- solve (H + mu I) dx = -g ; x_new = x + dx ; denom = dx.(mu*dx - g) ----
__global__ void icp_solve_kernel(float* __restrict__ ws) {
  if (threadIdx.x == 0) {
    float mu = ws[WS_MU];
    float M[6][7];
    for (int a = 0; a < 6; ++a) {
      for (int b = 0; b < 6; ++b) M[a][b] = ws[WS_HP + hidx(a, b)];
      M[a][a] += mu;
      M[a][6] = -ws[WS_G + a];
    }
    for (int col = 0; col < 6; ++col) {
      int piv = col;
      for (int r = col + 1; r < 6; ++r)
        if (fabsf(M[r][col]) > fabsf(M[piv][col])) piv = r;
      if (piv != col)
        for (int cc = 0; cc < 7; ++cc) { float tmp = M[col][cc]; M[col][cc] = M[piv][cc]; M[piv][cc] = tmp; }
      float inv = 1.0f / M[col][col];
      for (int r = col + 1; r < 6; ++r) {
        float f = M[r][col] * inv;
        for (int cc = col; cc < 7; ++cc) M[r][cc] -= f * M[col][cc];
      }
    }
    float dx[6];
    for (int a = 5; a >= 0; --a) {
      float s = M[a][6];
      for (int b = a + 1; b < 6; ++b) s -= M[a][b] * dx[b];
      dx[a] = s / M[a][a];
    }
    float denom = 0.0f;
    for (int a = 0; a < 6; ++a) {
      ws[WS_XNEW + a] = ws[WS_X + a] + dx[a];
      denom += dx[a] * (mu * dx[a] - ws[WS_G + a]);
    }
    ws[WS_DENOM] = denom;
  }
}

// ---- LM accept/reject ----
__global__ void icp_update_kernel(float* __restrict__ ws) {
  if (threadIdx.x == 0) {
    float F = ws[WS_F], Fn = ws[WS_FN];
    float denom = ws[WS_DENOM];
    float dn = (fabsf(denom) < 1e-12f) ? 1e-12f : denom;
    float rho = (F - Fn) / dn;
    float mu = ws[WS_MU], nu = ws[WS_NU];
    if (rho > 0.0f) {
      for (int k = 0; k < 6; ++k)  ws[WS_X + k] = ws[WS_XNEW + k];
      for (int k = 0; k < 21; ++k) ws[WS_HP + k] = ws[WS_HN + k];
      for (int k = 0; k < 6; ++k)  ws[WS_G + k] = ws[WS_GN + k];
      ws[WS_F] = Fn;
      float s = 2.0f * rho - 1.0f;
      float fac = 1.0f - s * s * s;
      ws[WS_MU] = mu * fmaxf(1.0f / 3.0f, fac);
      ws[WS_NU] = 2.0f;
    } else {
      ws[WS_MU] = mu * nu;
      ws[WS_NU] = 2.0f * nu;
    }
  }
}

// ---- finalize: out[0..15] = se3_exp(x) row-major, out[16] = F/N ----
__global__ void icp_finalize_kernel(const float* __restrict__ ws, float* __restrict__ out) {
  if (threadIdx.x == 0) {
    float R[9], t[3];
    se3_exp_dev(ws + WS_X, R, t);
    out[0]  = R[0]; out[1]  = R[1]; out[2]  = R[2]; out[3]  = t[0];
    out[4]  = R[3]; out[5]  = R[4]; out[6]  = R[5]; out[7]  = t[1];
    out[8]  = R[6]; out[9]  = R[7]; out[10] = R[8]; out[11] = t[2];
    out[12] = 0.0f; out[13] = 0.0f; out[14] = 0.0f; out[15] = 1.0f;
    out[16] = ws[WS_F] * (1.0f / (float)NPTS);
  }
}

extern "C" void kernel_launch(void* const* d_in, const int* in_sizes, int n_in,
                              void* d_out, int out_size, void* d_ws, size_t ws_size,
                              hipStream_t stream) {
  (void)in_sizes; (void)n_in; (void)out_size; (void)ws_size;
  const float* ref = (const float*)d_in[0];
  const float* trg = (const float*)d_in[1];
  const float* nrm = (const float*)d_in[2];
  float* ws  = (float*)d_ws;
  float* out = (float*)d_out;
  const int nred = NPTS / 256;          // residual/reduction blocks

#define ICP_EVAL()                                                            \
  do {                                                                        \
    icp_pose_kernel<<<1, 32, 0, stream>>>(ws);                                \
    icp_nn_kernel<<<NPTS / 128, 256, 0, stream>>>(ref, ws);                   \
    icp_resjac_partial<<<nred, 256, 0, stream>>>(ref, trg, nrm, ws);          \
    icp_resjac_final<<<1, 32, 0, stream>>>(ws, nred);                         \
  } while (0)

  icp_init_kernel<<<NPTS / 256, 256, 0, stream>>>(trg, ws);
  ICP_EVAL();                                  // evaluate at x = 0
  icp_lm_start<<<1, 32, 0, stream>>>(ws);
  for (int it = 0; it < 10; ++it) {
    icp_solve_kernel<<<1, 32, 0, stream>>>(ws);
    ICP_EVAL();                                // evaluate at x_new
    icp_update_kernel<<<1, 32, 0, stream>>>(ws);
  }
  icp_finalize_kernel<<<1, 32, 0, stream>>>(ws, out);
#undef ICP_EVAL
}